// PairModel_74672301408850
// MI455X (gfx1250) — compile-verified
//
#include <hip/hip_runtime.h>

// ---------------------------------------------------------------------------
// GL_Pair premlp fused pipeline for MI455X (gfx1250, wave32)
//
//  Stage 1 (WMMA f32):  ha[m,c] = sum_k pred_a[m,k]*W1[c,k]       + b1[c]
//                       hb[j,c] = sum_k pred_b[j,k]*W1[c,128+k]
//  Stage 2 (VALU):      out[m,j] = sum_c relu(ha[m,c]+hb[j,c])*w2[c] + b2
// ---------------------------------------------------------------------------

typedef float v2f __attribute__((ext_vector_type(2)));
typedef float v8f __attribute__((ext_vector_type(8)));

#define HIDDEN 128
#define TWOH   256

// ---------------------------------------------------------------------------
// Stage 1: one 16x16 f32 C-tile per wave via V_WMMA_F32_16X16X4_F32.
// A fragment (16x4 f32, 2 VGPRs): lane L, vgpr v  -> A[L%16][2*(L/16)+v]
// B fragment (4x16 f32, 2 VGPRs): lane L, vgpr v  -> B[2*(L/16)+v][L%16]
// C/D        (16x16 f32, 8 VGPRs): lane L, vgpr v -> C[v+8*(L/16)][L%16]
// ---------------------------------------------------------------------------
__global__ __launch_bounds__(256) void gl_pair_gemm_wmma(
    const float* __restrict__ pred_a,
    const float* __restrict__ pred_b,
    const float* __restrict__ W1,      // [TWOH, TWOH] row major
    const float* __restrict__ b1,      // [TWOH]
    float* __restrict__ ha,            // [N, TWOH]
    float* __restrict__ hb,            // [N, TWOH]
    int nMTiles)                       // N/16
{
    const int wave = threadIdx.x >> 5;
    const int lane = threadIdx.x & 31;
    const int half = lane >> 4;        // 0..1
    const int r    = lane & 15;        // 0..15

    int tile = blockIdx.x * 8 + wave;
    const int perMat = nMTiles * (TWOH / 16);
    const int which  = (tile >= perMat) ? 1 : 0;   // 0 = ha path, 1 = hb path
    tile -= which * perMat;
    const int tileM = tile >> 4;                   // TWOH/16 == 16 n-tiles
    const int tileN = tile & 15;

    const float* A    = which ? pred_b : pred_a;
    const int    koff = which ? HIDDEN : 0;

    const int arow = tileM * 16 + r;   // m index
    const int brow = tileN * 16 + r;   // n index == row of W1

    const float* aptr = A  + (size_t)arow * HIDDEN + 2 * half;
    const float* bptr = W1 + (size_t)brow * TWOH   + koff + 2 * half;

    v8f acc = {};
#pragma unroll 4
    for (int kk = 0; kk < HIDDEN; kk += 4) {
        v2f af = *(const v2f*)(aptr + kk);   // A[arow][kk + 2*half + {0,1}]
        v2f bf = *(const v2f*)(bptr + kk);   // W1[brow][koff + kk + 2*half + {0,1}]
        acc = __builtin_amdgcn_wmma_f32_16x16x4_f32(
            /*neg_a=*/false, af, /*neg_b=*/false, bf,
            /*c_mod=*/(short)0, acc, /*reuse_a=*/false, /*reuse_b=*/false);
    }

    float* D = which ? hb : ha;
    const int   col  = tileN * 16 + r;
    const float bias = which ? 0.0f : b1[col];
#pragma unroll
    for (int v = 0; v < 8; ++v) {
        const int row = tileM * 16 + v + 8 * half;
        D[(size_t)row * TWOH + col] = acc[v] + bias;
    }
}

// ---------------------------------------------------------------------------
// Stage 2: 64x64 output tile per 256-thread block, 4x4 micro-tile per thread.
// LDS holds transposed [k][m] chunks with stride 68 (padding -> conflict-free
// b128 reads, 16B aligned since 68*4 = 272 bytes).
// ---------------------------------------------------------------------------
#define KC   64
#define LDST 68

__global__ __launch_bounds__(256) void gl_pair_score(
    const float* __restrict__ ha,
    const float* __restrict__ hb,
    const float* __restrict__ w2,   // [TWOH]
    const float* __restrict__ b2,   // [1]
    float* __restrict__ out,        // [N, N]
    int N)
{
    __shared__ __align__(16) float sa[KC * LDST];
    __shared__ __align__(16) float sb[KC * LDST];
    __shared__ __align__(16) float sw[KC];

    const int tid = threadIdx.x;
    const int tx  = tid & 15;       // j micro-tile
    const int ty  = tid >> 4;       // m micro-tile
    const int m0  = blockIdx.y * 64;
    const int j0  = blockIdx.x * 64;

    // staging assignment: thread covers k = kq..kq+3 for rows mrow+16*rr
    const int kq   = (tid & 15) * 4;
    const int mrow = tid >> 4;

    float acc[4][4] = {};

    for (int k0 = 0; k0 < TWOH; k0 += KC) {
        // ---- stage chunk into LDS (transpose [m][k] -> [k][m]) ----
#pragma unroll
        for (int rr = 0; rr < 4; ++rr) {
            const int m = mrow + 16 * rr;
            float4 va = *(const float4*)&ha[(size_t)(m0 + m) * TWOH + k0 + kq];
            sa[(kq + 0) * LDST + m] = va.x;
            sa[(kq + 1) * LDST + m] = va.y;
            sa[(kq + 2) * LDST + m] = va.z;
            sa[(kq + 3) * LDST + m] = va.w;
            float4 vb = *(const float4*)&hb[(size_t)(j0 + m) * TWOH + k0 + kq];
            sb[(kq + 0) * LDST + m] = vb.x;
            sb[(kq + 1) * LDST + m] = vb.y;
            sb[(kq + 2) * LDST + m] = vb.z;
            sb[(kq + 3) * LDST + m] = vb.w;
        }
        if (tid < KC) sw[tid] = w2[k0 + tid];

        // prefetch next chunk while we compute (global_prefetch_b8)
        if (k0 + KC < TWOH) {
            __builtin_prefetch(&ha[(size_t)(m0 + mrow) * TWOH + k0 + KC + kq], 0, 0);
            __builtin_prefetch(&hb[(size_t)(j0 + mrow) * TWOH + k0 + KC + kq], 0, 0);
        }
        __syncthreads();

        // ---- compute: per k, 2x ds_load_b128 + 1 broadcast + 48 VALU ----
#pragma unroll 8
        for (int k = 0; k < KC; ++k) {
            float4 av = *(const float4*)&sa[k * LDST + 4 * ty];
            float4 bv = *(const float4*)&sb[k * LDST + 4 * tx];
            const float w = sw[k];
            const float a_[4] = {av.x, av.y, av.z, av.w};
            const float b_[4] = {bv.x, bv.y, bv.z, bv.w};
#pragma unroll
            for (int i = 0; i < 4; ++i)
#pragma unroll
                for (int j = 0; j < 4; ++j)
                    acc[i][j] = __builtin_fmaf(
                        fmaxf(a_[i] + b_[j], 0.0f), w, acc[i][j]);
        }
        __syncthreads();
    }

    const float bias2 = b2[0];
#pragma unroll
    for (int i = 0; i < 4; ++i) {
        float4 o;
        o.x = acc[i][0] + bias2;
        o.y = acc[i][1] + bias2;
        o.z = acc[i][2] + bias2;
        o.w = acc[i][3] + bias2;
        *(float4*)&out[(size_t)(m0 + 4 * ty + i) * N + j0 + 4 * tx] = o;
    }
}

// ---------------------------------------------------------------------------
// Launch wrapper. Inputs (setup_inputs order):
//   0: pred_a [N,128] f32     1: pred_b [N,128] f32
//   2: W1 [256,256] f32       3: b1 [256] f32
//   4: W2 [1,256] f32         5: b2 [1] f32
// Output: logits [N,N] f32.  Workspace: ha, hb ( 2*N*256 floats ).
// ---------------------------------------------------------------------------
extern "C" void kernel_launch(void* const* d_in, const int* in_sizes, int n_in,
                              void* d_out, int out_size, void* d_ws, size_t ws_size,
                              hipStream_t stream) {
    (void)n_in; (void)out_size; (void)ws_size;

    const float* pred_a = (const float*)d_in[0];
    const float* pred_b = (const float*)d_in[1];
    const float* W1     = (const float*)d_in[2];
    const float* b1     = (const float*)d_in[3];
    const float* w2     = (const float*)d_in[4];
    const float* b2     = (const float*)d_in[5];
    float*       out    = (float*)d_out;

    const int N = in_sizes[0] / HIDDEN;   // 1024

    float* ha = (float*)d_ws;
    float* hb = ha + (size_t)N * TWOH;

    // Stage 1: 2 * (N/16) * 16 waves, 8 waves (256 threads) per block
    const int nMTiles    = N / 16;
    const int totalWaves = 2 * nMTiles * (TWOH / 16);
    gl_pair_gemm_wmma<<<totalWaves / 8, 256, 0, stream>>>(
        pred_a, pred_b, W1, b1, ha, hb, nMTiles);

    // Stage 2: 64x64 tile per block
    dim3 grid(N / 64, N / 64);
    gl_pair_score<<<grid, 256, 0, stream>>>(ha, hb, w2, b2, out, N);
}